// GAT_70274254897513
// MI455X (gfx1250) — compile-verified
//
#include <hip/hip_runtime.h>
#include <hip/hip_bf16.h>

// ---------------------------------------------------------------------------
// GAT (3 layers) + inner-product decoder for MI455X (gfx1250, wave32, WMMA).
// All dense GEMMs use V_WMMA_F32_16X16X4_F32 (full f32 precision, matching the
// reference). N = 10000 = 625*16 tiles exactly -> no predication around WMMA,
// EXEC stays all-ones as the ISA requires.
// ---------------------------------------------------------------------------

typedef __attribute__((ext_vector_type(2))) float v2f;
typedef __attribute__((ext_vector_type(8))) float v8f;

#define GAT_N     10000
#define GAT_H     4
#define NEG_SLOPE 0.2f

// ordered-int key for -inf:  bits(-inf)=0xFF800000 (negative) -> ^0x7FFFFFFF
#define NEG_INF_KEY ((int)0x807FFFFF)

// ---- ordered-int encoding for float atomicMax (handles negatives) ----------
__device__ __forceinline__ int f2key(float f) {
  int b = __float_as_int(f);
  return b >= 0 ? b : (b ^ 0x7FFFFFFF);
}
__device__ __forceinline__ float key2f(int k) {
  int b = k >= 0 ? k : (k ^ 0x7FFFFFFF);
  return __int_as_float(b);
}

// ---------------------------------------------------------------------------
// C[N,M] = A[N,K] @ B[K,M].  One wave computes a 16x64 output tile with four
// f32 accumulators sharing the A fragment. N%16==0, M%64==0, K%4==0.
// ---------------------------------------------------------------------------
__global__ void gat_gemm_wmma_f32(const float* __restrict__ A,
                                  const float* __restrict__ B,
                                  float* __restrict__ C,
                                  int N, int K, int M) {
  int wave = (blockIdx.x * blockDim.x + threadIdx.x) >> 5;
  int lane = threadIdx.x & 31;
  int mt = M >> 6;                      // 64-wide column tiles
  int row_t = wave / mt;
  int col_t = wave - row_t * mt;
  if (row_t * 16 >= N) return;          // wave-uniform guard
  int rowBase = row_t * 16;
  int colBase = col_t * 64;

  int r   = lane & 15;
  int klo = (lane >> 4) << 1;           // 0 for lanes 0-15, 2 for lanes 16-31

  v8f acc0 = {}; v8f acc1 = {}; v8f acc2 = {}; v8f acc3 = {};

  const float* Arow = A + (size_t)(rowBase + r) * K;
  for (int k0 = 0; k0 < K; k0 += 4) {
    int ka = k0 + klo;
    v2f a; a.x = Arow[ka]; a.y = Arow[ka + 1];
    const float* Bp = B + (size_t)ka * M + colBase + r;
    v2f b0; b0.x = Bp[0];  b0.y = Bp[M];
    v2f b1; b1.x = Bp[16]; b1.y = Bp[M + 16];
    v2f b2; b2.x = Bp[32]; b2.y = Bp[M + 32];
    v2f b3; b3.x = Bp[48]; b3.y = Bp[M + 48];
    acc0 = __builtin_amdgcn_wmma_f32_16x16x4_f32(false, a, false, b0, (short)0, acc0, false, false);
    acc1 = __builtin_amdgcn_wmma_f32_16x16x4_f32(false, a, false, b1, (short)0, acc1, false, false);
    acc2 = __builtin_amdgcn_wmma_f32_16x16x4_f32(false, a, false, b2, (short)0, acc2, false, false);
    acc3 = __builtin_amdgcn_wmma_f32_16x16x4_f32(false, a, false, b3, (short)0, acc3, false, false);
  }

  // C/D layout: VGPR v -> row v (lanes 0-15) / row v+8 (lanes 16-31), col=lane&15
  int cr = rowBase + ((lane >> 4) << 3);
  int cc = lane & 15;
#pragma unroll
  for (int v = 0; v < 8; ++v) {
    float* Cp = C + (size_t)(cr + v) * M + colBase + cc;
    Cp[0]  = acc0[v];
    Cp[16] = acc1[v];
    Cp[32] = acc2[v];
    Cp[48] = acc3[v];
  }
}

// ---------------------------------------------------------------------------
// adj[N,N] = L @ L^T, L = logits [N,64]. One wave per 16x16 tile, K=64.
// B fragment of tile col block is just L rows colBase..colBase+15.
// ---------------------------------------------------------------------------
__global__ void gat_adj_wmma_f32(const float* __restrict__ L,
                                 float* __restrict__ C,
                                 int NT, int N) {
  int wave = (blockIdx.x * blockDim.x + threadIdx.x) >> 5;
  int lane = threadIdx.x & 31;
  int row_t = wave / NT;
  int col_t = wave - row_t * NT;
  if (row_t >= NT) return;              // wave-uniform guard
  int rowBase = row_t * 16;
  int colBase = col_t * 16;

  int r   = lane & 15;
  int klo = (lane >> 4) << 1;

  const float* Ap = L + (size_t)(rowBase + r) * 64;
  const float* Bp = L + (size_t)(colBase + r) * 64;

  v8f acc = {};
#pragma unroll
  for (int k0 = 0; k0 < 64; k0 += 4) {
    int ka = k0 + klo;
    v2f a; a.x = Ap[ka]; a.y = Ap[ka + 1];
    v2f b; b.x = Bp[ka]; b.y = Bp[ka + 1];
    acc = __builtin_amdgcn_wmma_f32_16x16x4_f32(false, a, false, b, (short)0, acc, false, false);
  }

  int cr = rowBase + ((lane >> 4) << 3);
  int cc = colBase + (lane & 15);
#pragma unroll
  for (int v = 0; v < 8; ++v)
    C[(size_t)(cr + v) * N + cc] = acc[v];
}

// ---------------------------------------------------------------------------
// el[n,h] = sum_d z[n,h,d]*al[h,d];  er likewise. One wave per (n,h).
// ---------------------------------------------------------------------------
__global__ void gat_scores(const float* __restrict__ z,
                           const float* __restrict__ al,
                           const float* __restrict__ ar,
                           float* __restrict__ el, float* __restrict__ er,
                           int NH, int H, int D) {
  int wave = (blockIdx.x * blockDim.x + threadIdx.x) >> 5;
  int lane = threadIdx.x & 31;
  if (wave >= NH) return;
  int h = wave % H;
  const float* zp  = z + (size_t)wave * D;
  const float* alp = al + h * D;
  const float* arp = ar + h * D;
  float sl = 0.f, sr = 0.f;
  for (int d = lane; d < D; d += 32) {
    float zv = zp[d];
    sl += zv * alp[d];
    sr += zv * arp[d];
  }
#pragma unroll
  for (int off = 16; off > 0; off >>= 1) {
    sl += __shfl_down(sl, off, 32);
    sr += __shfl_down(sr, off, 32);
  }
  if (lane == 0) { el[wave] = sl; er[wave] = sr; }
}

__global__ void gat_init_keys(int* __restrict__ p, int n, int key) {
  int t = blockIdx.x * blockDim.x + threadIdx.x;
  if (t < n) p[t] = key;
}

// ---------------------------------------------------------------------------
// Pass 1: segment max of leakyrelu(el[src]+er[dst]) over dst.
// ---------------------------------------------------------------------------
__global__ void gat_edge_max(const long long* __restrict__ src,
                             const long long* __restrict__ dst,
                             const float* __restrict__ el,
                             const float* __restrict__ er,
                             int* __restrict__ emax, int E, int H) {
  int t = blockIdx.x * blockDim.x + threadIdx.x;
  if (t >= E * H) return;
  int e = t / H, h = t - e * H;
  int s = (int)src[e], d = (int)dst[e];
  float v = el[s * H + h] + er[d * H + h];
  v = v >= 0.f ? v : NEG_SLOPE * v;
  atomicMax(&emax[d * H + h], f2key(v));
}

// ---------------------------------------------------------------------------
// Pass 2: w = exp(e - emax[dst]); denom[dst] += w; store w per edge.
// ---------------------------------------------------------------------------
__global__ void gat_edge_exp(const long long* __restrict__ src,
                             const long long* __restrict__ dst,
                             const float* __restrict__ el,
                             const float* __restrict__ er,
                             const int* __restrict__ emax,
                             float* __restrict__ ew,
                             float* __restrict__ denom, int E, int H) {
  int t = blockIdx.x * blockDim.x + threadIdx.x;
  if (t >= E * H) return;
  int e = t / H, h = t - e * H;
  int s = (int)src[e], d = (int)dst[e];
  float v = el[s * H + h] + er[d * H + h];
  v = v >= 0.f ? v : NEG_SLOPE * v;
  float w = expf(v - key2f(emax[d * H + h]));
  ew[t] = w;
  atomicAdd(&denom[d * H + h], w);
}

// ---------------------------------------------------------------------------
// Pass 3: acc[dst,h,:] += w * z[src,h,:].  One 256-thread block per edge.
// ---------------------------------------------------------------------------
__global__ void gat_edge_agg(const long long* __restrict__ src,
                             const long long* __restrict__ dst,
                             const float* __restrict__ z,
                             const float* __restrict__ ew,
                             float* __restrict__ acc,
                             int E, int H, int HD, int dshift) {
  int e = blockIdx.x;
  if (e >= E) return;
  int s = (int)src[e], d = (int)dst[e];
  const float* zp = z + (size_t)s * HD;
  float* ap = acc + (size_t)d * HD;
  const float* wp = ew + (size_t)e * H;
  for (int i = threadIdx.x; i < HD; i += blockDim.x)
    atomicAdd(&ap[i], wp[i >> dshift] * zp[i]);
}

// out = relu(acc/denom + bias), in place.
__global__ void gat_finalize_relu(float* __restrict__ acc,
                                  const float* __restrict__ denom,
                                  const float* __restrict__ bias,
                                  int total, int H, int D) {
  int t = blockIdx.x * blockDim.x + threadIdx.x;
  if (t >= total) return;
  int d  = t % D;
  int nh = t / D;
  int h  = nh % H;
  float v = acc[t] / denom[nh] + bias[h * D + d];
  acc[t] = v > 0.f ? v : 0.f;
}

// logits[n,d] = mean_h(acc[n,h,d]/denom[n,h] + bias[h,d])   (no relu)
__global__ void gat_finalize_mean(const float* __restrict__ acc,
                                  const float* __restrict__ denom,
                                  const float* __restrict__ bias,
                                  float* __restrict__ logits,
                                  int N, int H, int D) {
  int t = blockIdx.x * blockDim.x + threadIdx.x;
  if (t >= N * D) return;
  int n = t / D, d = t - n * D;
  float s = 0.f;
#pragma unroll
  for (int h = 0; h < GAT_H; ++h)
    s += acc[((size_t)(n * H + h)) * D + d] / denom[n * H + h] + bias[h * D + d];
  logits[t] = s * (1.0f / GAT_H);
}

// ---------------------------------------------------------------------------
extern "C" void kernel_launch(void* const* d_in, const int* in_sizes, int n_in,
                              void* d_out, int out_size, void* d_ws, size_t ws_size,
                              hipStream_t stream) {
  const float*     x    = (const float*)d_in[0];
  const long long* esrc = (const long long*)d_in[1];
  const long long* edst = (const long long*)d_in[2];
  const float* W0  = (const float*)d_in[3];
  const float* al0 = (const float*)d_in[4];
  const float* ar0 = (const float*)d_in[5];
  const float* b0  = (const float*)d_in[6];
  const float* W1  = (const float*)d_in[7];
  const float* al1 = (const float*)d_in[8];
  const float* ar1 = (const float*)d_in[9];
  const float* b1  = (const float*)d_in[10];
  const float* W2  = (const float*)d_in[11];
  const float* al2 = (const float*)d_in[12];
  const float* ar2 = (const float*)d_in[13];
  const float* b2  = (const float*)d_in[14];

  const int N = GAT_N, H = GAT_H;
  const int E = in_sizes[1];            // 320000 (310000 random + N self loops)

  // workspace layout (~45 MB total)
  char* ws = (char*)d_ws;
  float* bufA  = (float*)(ws);                      // z buffer, 20 MB
  float* bufB  = (float*)(ws + 20480000);           // h / accumulator, 20 MB
  float* ew    = (float*)(ws + 40960000);           // per-edge weights, 5.12 MB
  float* el    = (float*)(ws + 46080000);           // [N,H]
  float* er    = (float*)(ws + 46240000);
  int*   emax  = (int*)  (ws + 46400000);
  float* denom = (float*)(ws + 46560000);

  float* adj    = (float*)d_out;
  float* logits = (float*)d_out + (size_t)N * N;    // reference returns (adj, logits)

  (void)ws_size; (void)n_in; (void)out_size;

  auto layer = [&](const float* hin, int K, const float* W, const float* al_,
                   const float* ar_, const float* bias, int D, bool last) {
    int M = H * D;
    // 1) z = hin @ W
    int gwaves = (N / 16) * (M / 64);
    gat_gemm_wmma_f32<<<(gwaves + 7) / 8, 256, 0, stream>>>(hin, W, bufA, N, K, M);
    // 2) attention scores
    int NH = N * H;
    gat_scores<<<(NH * 32 + 255) / 256, 256, 0, stream>>>(bufA, al_, ar_, el, er, NH, H, D);
    // 3) init reductions
    (void)hipMemsetAsync(denom, 0, (size_t)NH * 4, stream);
    (void)hipMemsetAsync(bufB, 0, (size_t)N * M * 4, stream);
    gat_init_keys<<<(NH + 255) / 256, 256, 0, stream>>>(emax, NH, NEG_INF_KEY);
    // 4) edge softmax
    int EH = E * H;
    gat_edge_max<<<(EH + 255) / 256, 256, 0, stream>>>(esrc, edst, el, er, emax, E, H);
    gat_edge_exp<<<(EH + 255) / 256, 256, 0, stream>>>(esrc, edst, el, er, emax, ew, denom, E, H);
    // 5) message aggregation
    int dshift = (D == 128) ? 7 : 6;
    gat_edge_agg<<<E, 256, 0, stream>>>(esrc, edst, bufA, ew, bufB, E, H, M, dshift);
    // 6) finalize
    if (!last) {
      gat_finalize_relu<<<((N * M) + 255) / 256, 256, 0, stream>>>(bufB, denom, bias, N * M, H, D);
    } else {
      gat_finalize_mean<<<((N * D) + 255) / 256, 256, 0, stream>>>(bufB, denom, bias, logits, N, H, D);
    }
  };

  layer(x,    256, W0, al0, ar0, b0, 128, false);   // h0 -> bufB
  layer(bufB, 512, W1, al1, ar1, b1, 128, false);   // h1 -> bufB (gemm reads it first)
  layer(bufB, 512, W2, al2, ar2, b2,  64, true);    // logits -> d_out tail

  // adj = logits @ logits^T  (400 MB store; bandwidth-bound)
  int NT = N / 16;                                  // 625
  long long twaves = (long long)NT * NT;            // 390625
  gat_adj_wmma_f32<<<(unsigned)((twaves + 7) / 8), 256, 0, stream>>>(logits, adj, NT, N);
}